// Model_70471823392997
// MI455X (gfx1250) — compile-verified
//
#include <hip/hip_runtime.h>

// ---- problem constants (from the reference) --------------------------------
#define NUM_TOKENS 16384
#define NUM_HEADS  8
#define HEAD_SIZE  128
#define NUM_BLOCKS 4096
#define BLOCK_SIZE 16
#define NUM_SLOTS  (NUM_BLOCKS * BLOCK_SIZE)          // 65536
#define ROW_ELEMS  (NUM_HEADS * HEAD_SIZE)            // 1024 floats = 4 KiB/row

typedef float __attribute__((ext_vector_type(4))) f4;
// Exact pointee type of the async-LDS builtins per hipcc's diagnostic:
//   '__attribute__((__vector_size__(4 * sizeof(int)))) int __device__ *'
typedef int i4v __attribute__((vector_size(16)));

// Detect the CDNA5 async global<->LDS builtins (ASYNCcnt data-mover path).
#if defined(__HIP_DEVICE_COMPILE__) && defined(__gfx1250__) && \
    __has_builtin(__builtin_amdgcn_global_load_async_to_lds_b128) && \
    __has_builtin(__builtin_amdgcn_global_store_async_from_lds_b128)
#define USE_ASYNC_LDS 1
#else
#define USE_ASYNC_LDS 0
#endif

#if defined(__HIP_DEVICE_COMPILE__)
__device__ __forceinline__ void wait_asynccnt0() {
#if __has_builtin(__builtin_amdgcn_s_wait_asynccnt)
    __builtin_amdgcn_s_wait_asynccnt(0);
#else
    asm volatile("s_wait_asynccnt 0" ::: "memory");
#endif
}
#endif

// ---------------------------------------------------------------------------
// Table path (preferred): slot -> token inverse map in workspace, then a
// single gather pass that touches each byte of the 512 MiB output exactly once
// and each source byte exactly once.  ~1.0 GiB total traffic.
// ---------------------------------------------------------------------------

__global__ void kv_init_table(int* __restrict__ table) {
    unsigned i = blockIdx.x * blockDim.x + threadIdx.x;
    if (i < NUM_SLOTS) table[i] = -1;
}

__global__ void kv_fill_table(const int* __restrict__ slot_mapping,
                              int* __restrict__ table) {
    unsigned i = blockIdx.x * blockDim.x + threadIdx.x;
    if (i < NUM_TOKENS) {
        int s = slot_mapping[i];
        if (s >= 0 && s < NUM_SLOTS) table[s] = (int)i;
    }
}

// grid = 2*NUM_SLOTS blocks of 256 threads; one block = one 4 KiB slot row.
// Block index < NUM_SLOTS -> key cache half, else value cache half.
__global__ void kv_gather_rows(const float* __restrict__ key,
                               const float* __restrict__ value,
                               const float* __restrict__ key_cache,
                               const float* __restrict__ value_cache,
                               const int* __restrict__ table,
                               float* __restrict__ out) {
    const unsigned bid  = blockIdx.x;
    const unsigned slot = bid & (NUM_SLOTS - 1);
    const bool is_v     = bid >= NUM_SLOTS;

    const int tok = table[slot];                      // uniform per block
    const float* src;
    if (tok >= 0) {
        src = (is_v ? value : key) + (size_t)tok * ROW_ELEMS;
    } else {
        src = (is_v ? value_cache : key_cache) + (size_t)slot * ROW_ELEMS;
    }
    float* dst = out + (is_v ? (size_t)NUM_SLOTS * ROW_ELEMS : (size_t)0)
                     + (size_t)slot * ROW_ELEMS;

    const unsigned e = threadIdx.x * 4u;              // 256 thr * float4 = row

#if USE_ASYNC_LDS
    // CDNA5 async data-mover: global -> LDS -> global, 16B per lane, no VGPR
    // data movement.  Each lane only touches its own LDS bytes, so a wave-level
    // s_wait_asynccnt between the load and the store is sufficient (async loads
    // retire in order; ASYNCcnt==0 => data landed in LDS).
    __shared__ __attribute__((aligned(16))) float lds_buf[ROW_ELEMS];
    __builtin_amdgcn_global_load_async_to_lds_b128(
        (__attribute__((address_space(1))) i4v*)(void*)(src + e),
        (__attribute__((address_space(3))) i4v*)(void*)(lds_buf + e),
        0, 0);
    wait_asynccnt0();
    __builtin_amdgcn_global_store_async_from_lds_b128(
        (__attribute__((address_space(1))) i4v*)(void*)(dst + e),
        (__attribute__((address_space(3))) i4v*)(void*)(lds_buf + e),
        0, 0);
    // S_ENDPGM performs an implicit wait-idle, draining the async stores.
#else
    // Fallback: 128-bit non-temporal streaming copy.  NT hints keep the
    // 512 MiB one-shot output stream from thrashing the 192 MB L2.
    f4 v = __builtin_nontemporal_load((const f4*)(src + e));
    __builtin_nontemporal_store(v, (f4*)(dst + e));
#endif
}

// ---------------------------------------------------------------------------
// No-workspace fallback: full cache copy then scatter (~1.28 GiB traffic).
// ---------------------------------------------------------------------------

__global__ void kv_copy_cache(const float* __restrict__ key_cache,
                              const float* __restrict__ value_cache,
                              float* __restrict__ out) {
    const size_t half = (size_t)NUM_SLOTS * ROW_ELEMS / 4;   // f4 elements
    size_t i = (size_t)blockIdx.x * blockDim.x + threadIdx.x;
    const f4* src = (i < half) ? ((const f4*)key_cache + i)
                               : ((const f4*)value_cache + (i - half));
    f4 v = __builtin_nontemporal_load(src);
    __builtin_nontemporal_store(v, (f4*)out + i);
}

__global__ void kv_scatter_tokens(const float* __restrict__ key,
                                  const float* __restrict__ value,
                                  const int* __restrict__ slot_mapping,
                                  float* __restrict__ out) {
    const unsigned bid  = blockIdx.x;
    const bool is_v     = bid >= NUM_TOKENS;
    const unsigned tok  = is_v ? bid - NUM_TOKENS : bid;
    const int slot = slot_mapping[tok];
    if (slot < 0 || slot >= NUM_SLOTS) return;               // drop semantics
    const float* src = (is_v ? value : key) + (size_t)tok * ROW_ELEMS;
    float* dst = out + (is_v ? (size_t)NUM_SLOTS * ROW_ELEMS : (size_t)0)
                     + (size_t)slot * ROW_ELEMS;
    const unsigned e = threadIdx.x * 4u;
    f4 v = *(const f4*)(src + e);
    __builtin_nontemporal_store(v, (f4*)(dst + e));
}

// ---------------------------------------------------------------------------

extern "C" void kernel_launch(void* const* d_in, const int* in_sizes, int n_in,
                              void* d_out, int out_size, void* d_ws, size_t ws_size,
                              hipStream_t stream) {
    const float* key          = (const float*)d_in[0];
    const float* value        = (const float*)d_in[1];
    const float* key_cache    = (const float*)d_in[2];
    const float* value_cache  = (const float*)d_in[3];
    const int*   slot_mapping = (const int*)  d_in[4];
    // d_in[5]/d_in[6] (k_scale/v_scale) are unused: kv_cache_dtype='auto'.
    float* out = (float*)d_out;

    (void)in_sizes; (void)n_in; (void)out_size;

    const int threads = 256;

    if (ws_size >= (size_t)NUM_SLOTS * sizeof(int)) {
        int* table = (int*)d_ws;
        kv_init_table<<<NUM_SLOTS / threads, threads, 0, stream>>>(table);
        kv_fill_table<<<NUM_TOKENS / threads, threads, 0, stream>>>(slot_mapping, table);
        kv_gather_rows<<<2 * NUM_SLOTS, threads, 0, stream>>>(
            key, value, key_cache, value_cache, table, out);
    } else {
        const size_t total_f4 = (size_t)2 * NUM_SLOTS * ROW_ELEMS / 4;
        kv_copy_cache<<<(unsigned)(total_f4 / threads), threads, 0, stream>>>(
            key_cache, value_cache, out);
        kv_scatter_tokens<<<2 * NUM_TOKENS, threads, 0, stream>>>(
            key, value, slot_mapping, out);
    }
}